// KANSpline1D_43696997269764
// MI455X (gfx1250) — compile-verified
//
#include <hip/hip_runtime.h>

#define BK 16     // number of basis functions K
#define PP 3      // spline degree
#define NSPAN 13  // non-degenerate knot spans for K=16, p=3
#define CCH 128   // channels

typedef __attribute__((ext_vector_type(2))) float v2f;
typedef __attribute__((ext_vector_type(8))) float v8f;

// ---------- compile-time basis-polynomial table ----------
// Clamped open-uniform knot vector on [-1,1]: 4x(-1), 12 interior, 4x(+1).
constexpr double knotc(int i) {
  return (i <= 3) ? -1.0 : ((i >= 16) ? 1.0 : -1.0 + 2.0 * (double)(i - 3) / 13.0);
}

struct BPTable {
  // c[j][m][pow]: coefficient of u^pow of basis N_{j+m} restricted to span j
  float c[NSPAN][4][4];
};

constexpr BPTable make_bp() {
  BPTable t{};
  for (int j = 0; j < NSPAN; ++j) {
    const int s = j + 3;
    double P[5][4] = {};
    P[3][0] = 1.0;  // N_s^0 = 1 on this span
    for (int r = 1; r <= PP; ++r) {
      double Q[5][4] = {};
      for (int i = s - r; i <= s; ++i) {
        const int li = i - (s - 3);
        double acc[4] = {};
        const double d1 = knotc(i + r) - knotc(i);
        if (d1 != 0.0) {  // (u - t_i)/d1 * P[li]
          const double ai = 1.0 / d1, bi = -knotc(i) / d1;
          for (int kp = 3; kp >= 0; --kp) {
            double v = bi * P[li][kp];
            if (kp > 0) v += ai * P[li][kp - 1];
            acc[kp] += v;
          }
        }
        const double d2 = knotc(i + r + 1) - knotc(i + 1);
        if (d2 != 0.0) {  // (t_{i+r+1} - u)/d2 * P[li+1]
          const double ai = -1.0 / d2, bi = knotc(i + r + 1) / d2;
          for (int kp = 3; kp >= 0; --kp) {
            double v = bi * P[li + 1][kp];
            if (kp > 0) v += ai * P[li + 1][kp - 1];
            acc[kp] += v;
          }
        }
        for (int kp = 0; kp < 4; ++kp) Q[li][kp] = acc[kp];
      }
      for (int a = 0; a < 5; ++a)
        for (int b = 0; b < 4; ++b) P[a][b] = Q[a][b];
    }
    for (int m = 0; m < 4; ++m)
      for (int kp = 0; kp < 4; ++kp) t.c[j][m][kp] = (float)P[m][kp];
  }
  return t;
}

constexpr BPTable kBP = make_bp();  // constexpr => usable from device, lives in .rodata

// ---------- setup: combine alpha with basis polys via WMMA ----------
// One wave per (16-channel tile, span):
//   D[M,n] = sum_k alpha[c0+M, j+k] * bp[j][k][n]   via V_WMMA_F32_16X16X4_F32
__global__ __launch_bounds__(32) void kan_setup_coeffs(const float* __restrict__ alpha,
                                                       float* __restrict__ Cw) {
  const int wave = (int)blockIdx.x;        // 0 .. (CCH/16)*NSPAN - 1
  const int j = wave % NSPAN;              // span index
  const int c0 = (wave / NSPAN) * 16;      // channel tile base

  const int lane = (int)(threadIdx.x & 31);
  const int half = lane >> 4;  // K-half per ISA 16x4 f32 A layout
  const int m16 = lane & 15;

  // A (16x4 f32): lanes 0-15: M rows, K={0,1}; lanes 16-31: same M, K={2,3}
  v2f A;
  A.x = alpha[(c0 + m16) * BK + (j + (half ? 2 : 0))];
  A.y = alpha[(c0 + m16) * BK + (j + (half ? 3 : 1))];
  // B (4x16 f32): row k striped across lanes; N = power index (only n<4 used)
  const int k0 = half ? 2 : 0;
  const int k1 = half ? 3 : 1;
  v2f Bv;
  Bv.x = (m16 < 4) ? kBP.c[j][k0][m16] : 0.0f;
  Bv.y = (m16 < 4) ? kBP.c[j][k1][m16] : 0.0f;

  v8f acc = {};
  acc = __builtin_amdgcn_wmma_f32_16x16x4_f32(false, A, false, Bv, (short)0, acc,
                                              false, false);

  // D layout: VGPR v -> lanes 0-15: (M=v, N=lane), lanes 16-31: (M=v+8, N=lane-16)
#pragma unroll
  for (int v = 0; v < 8; ++v) {
    const int M = v + (half ? 8 : 0);
    if (m16 < 4) Cw[(((c0 + M) * NSPAN) + j) * 4 + m16] = acc[v];
  }
}

// ---------- main: HBM-bound streaming ----------
// 1024 contiguous elements per block => channel uniform per block (H*W = 4096).
// b128 load -> affine -> span select -> one cubic (3 FMA) -> b128 store.
__global__ __launch_bounds__(256) void kan_apply(const float* __restrict__ x,
                                                 const float* __restrict__ a,
                                                 const float* __restrict__ b,
                                                 const float* __restrict__ g,
                                                 const float* __restrict__ bias,
                                                 const float* __restrict__ Cw,
                                                 float* __restrict__ out) {
  __shared__ float4 sc[NSPAN];
  const int blk = (int)blockIdx.x;
  const int c = (blk >> 2) & (CCH - 1);  // 4 blocks per (batch, channel) image
  if (threadIdx.x < NSPAN)
    sc[threadIdx.x] = reinterpret_cast<const float4*>(Cw)[c * NSPAN + (int)threadIdx.x];
  const float ac = a[c], bc = b[c], gc = g[c], oc = bias[c];
  __syncthreads();

  const long base = (long)blk * 1024 + (long)threadIdx.x * 4;
  __builtin_prefetch(x + base + 4096, 0, 0);  // global_prefetch_b8 on the stream
  const float4 xv = *reinterpret_cast<const float4*>(x + base);

  float xs[4] = {xv.x, xv.y, xv.z, xv.w};
  float os[4];
#pragma unroll
  for (int e = 0; e < 4; ++e) {
    const float xi = xs[e];
    const float u = fmaf(xi, ac, bc);  // clip(±1.5) only matters via |u|<=1 test
    float sp = 0.0f;
    if (u >= -1.0f && u <= 1.0f) {
      int jj = (int)((u + 1.0f) * 6.5f);  // 13 spans over [-1,1]
      jj = jj > 12 ? 12 : jj;             // u == 1 -> span 12 (left-limit correct)
      const float4 cf = sc[jj];
      sp = fmaf(fmaf(fmaf(cf.w, u, cf.z), u, cf.y), u, cf.x);
    }
    os[e] = fmaf(gc, xi, sp + oc);
  }
  float4 ov;
  ov.x = os[0]; ov.y = os[1]; ov.z = os[2]; ov.w = os[3];
  *reinterpret_cast<float4*>(out + base) = ov;
}

extern "C" void kernel_launch(void* const* d_in, const int* in_sizes, int n_in,
                              void* d_out, int out_size, void* d_ws, size_t ws_size,
                              hipStream_t stream) {
  const float* x     = (const float*)d_in[0];
  const float* a     = (const float*)d_in[1];
  const float* b     = (const float*)d_in[2];
  const float* alpha = (const float*)d_in[3];
  const float* g     = (const float*)d_in[4];
  const float* bias  = (const float*)d_in[5];
  float* Cw = (float*)d_ws;  // CCH*NSPAN*4 floats = 26,624 bytes of scratch

  kan_setup_coeffs<<<(CCH / 16) * NSPAN, 32, 0, stream>>>(alpha, Cw);

  const int blocks = out_size / 1024;  // 8192 blocks of 256 threads x 4 elems
  kan_apply<<<blocks, 256, 0, stream>>>(x, a, b, g, bias, Cw, (float*)d_out);
}